// MaskLinear_12524124635918
// MI455X (gfx1250) — compile-verified
//
#include <hip/hip_runtime.h>
#include <stdint.h>

typedef __attribute__((ext_vector_type(2))) float v2f;
typedef __attribute__((ext_vector_type(8))) float v8f;

#define B_DIM   64
#define T_DIM   32
#define J_DIM   128
#define N_DIM   64
#define L_DIM   256
#define INF_DIM 4096

#if __has_builtin(__builtin_amdgcn_global_load_async_to_lds_b128)
#define HAVE_ASYNC_LDS 1
typedef int v4i_vs __attribute__((vector_size(4 * sizeof(int))));
typedef __attribute__((address_space(1))) v4i_vs as1_v4i;   // global (prints __device__)
typedef __attribute__((address_space(3))) v4i_vs as3_v4i;   // LDS    (prints __shared__)
#endif

// ---------------------------------------------------------------------------
// Kernel 1: transpose full_weight (OUT_F x IN_F) -> wT (IN_F x OUT_F) so the
// per-(b,t) column gather becomes contiguous 1KB row copies (L2 resident).
// ---------------------------------------------------------------------------
__global__ __launch_bounds__(256) void weight_transpose_kernel(
    const float* __restrict__ w, float* __restrict__ wT)
{
    __shared__ float tile[32][33];
    const int bx = blockIdx.x;   // IN_F/32  = 128 tiles
    const int by = blockIdx.y;   // OUT_F/32 = 8 tiles
    const int tx = threadIdx.x;  // 32
    const int ty = threadIdx.y;  // 8

#pragma unroll
    for (int k = 0; k < 4; ++k) {
        const int o = by * 32 + ty + k * 8;       // row in w (OUT_F)
        const int i = bx * 32 + tx;               // col in w (IN_F)
        tile[ty + k * 8][tx] = w[(size_t)o * INF_DIM + i];
    }
    __syncthreads();
#pragma unroll
    for (int k = 0; k < 4; ++k) {
        const int i = bx * 32 + ty + k * 8;       // row in wT (IN_F)
        const int o = by * 32 + tx;               // col in wT (OUT_F)
        wT[(size_t)i * L_DIM + o] = tile[tx][ty + k * 8];
    }
}

// ---------------------------------------------------------------------------
// Kernel 2: one workgroup per (b,t): y[bt] = A(128x64) x B(64x256) via
// v_wmma_f32_16x16x4_f32.  B gathered into LDS with async-to-LDS copies
// (XOR-swizzled), A streamed from global in WMMA fragment layout.
// ---------------------------------------------------------------------------
__global__ __launch_bounds__(256) void mask_linear_wmma_kernel(
    const float* __restrict__ input,    // (B,T,J,N) fp32
    const int*   __restrict__ indices,  // (B,N,T)
    const float* __restrict__ wT,       // (IN_F, OUT_F) fp32
    float*       __restrict__ out)      // (B,T,J,L) fp32
{
    __shared__ float Bs[N_DIM * L_DIM]; // 64 x 256 fp32 = 64 KB, swizzled cols

    const int bt  = blockIdx.x;         // b*32 + t
    const int b   = bt >> 5;
    const int t   = bt & 31;
    const int tid = threadIdx.x;

    // ---- Gather B tile: Bs[n][l] = full_weight[l, indices[b,n,t]] = wT[idx][l]
    {
        const int n   = tid >> 2;                       // 4 threads per row n
        const int idx = indices[((size_t)b * N_DIM + n) * T_DIM + t];
        const float4* src = (const float4*)(wT + (size_t)idx * L_DIM);
        float4*       dst = (float4*)Bs + n * (L_DIM / 4);
        const int sw4 = ((n >> 1) & 1) << 2;            // XOR swizzle, float4 units
#pragma unroll
        for (int i = 0; i < 16; ++i) {
            const int l4 = (tid & 3) * 16 + i;
#if defined(HAVE_ASYNC_LDS)
            // Direct memory -> LDS, ASYNCcnt-tracked, no VGPR staging.
            // AS3 pointer value is the low 32 bits of the flat LDS address.
            __builtin_amdgcn_global_load_async_to_lds_b128(
                (as1_v4i*)(uintptr_t)(src + l4),
                (as3_v4i*)(uintptr_t)(uint32_t)(uintptr_t)(dst + (l4 ^ sw4)),
                0, 0);
#else
            dst[l4 ^ sw4] = src[l4];
#endif
        }
    }
#if defined(HAVE_ASYNC_LDS)
    asm volatile("s_wait_asynccnt 0x0" ::: "memory");
#endif
    __syncthreads();

    // ---- WMMA compute: wave w owns output rows [16w, 16w+16) x all 256 cols
    const int wave  = tid >> 5;
    const int lane  = tid & 31;
    const int l16   = lane & 15;
    const int lhalf = lane >> 4;
    const int m0    = wave * 16;

    // A fragment (16x4 f32): lane<16 -> (M=l16, K=kc..kc+1); lane>=16 -> K=kc+2..kc+3
    const float* Aptr = input + (size_t)bt * (J_DIM * N_DIM)
                              + (size_t)(m0 + l16) * N_DIM + 2 * lhalf;

    v8f acc[16] = {};
    const int colsw = lhalf << 4;   // swizzle for B reads: rows k,k+1 vs k+2,k+3

#pragma unroll
    for (int kc = 0; kc < N_DIM; kc += 4) {
        const v2f a = *(const v2f*)(Aptr + kc);
        const int row = kc + 2 * lhalf;
        const float* brow0 = Bs + row * L_DIM;
        const float* brow1 = brow0 + L_DIM;
#pragma unroll
        for (int nt = 0; nt < 16; ++nt) {
            const int colp = (nt * 16 + l16) ^ colsw;   // bank-conflict-free
            v2f bb;
            bb.x = brow0[colp];
            bb.y = brow1[colp];
            acc[nt] = __builtin_amdgcn_wmma_f32_16x16x4_f32(
                false, a, false, bb, (short)0, acc[nt], false, false);
        }
    }

    // ---- Store D: VGPR i holds (M = m0 + 8*lhalf + i, N = nt*16 + l16)
    float* outB = out + (size_t)bt * (J_DIM * L_DIM);
#pragma unroll
    for (int nt = 0; nt < 16; ++nt) {
#pragma unroll
        for (int i = 0; i < 8; ++i) {
            const int j = m0 + 8 * lhalf + i;
            outB[(size_t)j * L_DIM + nt * 16 + l16] = acc[nt][i];
        }
    }
}

// ---------------------------------------------------------------------------
extern "C" void kernel_launch(void* const* d_in, const int* in_sizes, int n_in,
                              void* d_out, int out_size, void* d_ws, size_t ws_size,
                              hipStream_t stream)
{
    const float* input   = (const float*)d_in[0];   // (B,T,J,N) fp32
    const int*   indices = (const int*)  d_in[1];   // (B,N,T)
    const float* weight  = (const float*)d_in[2];   // (OUT_F, IN_F) fp32
    float*       out     = (float*)d_out;
    float*       wT      = (float*)d_ws;            // IN_F x OUT_F = 4 MB

    weight_transpose_kernel<<<dim3(INF_DIM / 32, 256 / 32), dim3(32, 8), 0, stream>>>(
        weight, wT);

    mask_linear_wmma_kernel<<<B_DIM * T_DIM, 256, 0, stream>>>(
        input, indices, wT, out);
}